// FCOS_52544629899672
// MI455X (gfx1250) — compile-verified
//
#include <hip/hip_runtime.h>
#include <stdint.h>

#define NUM_CLASSES 80
#define N_POINTS    262144
#define MAX_DET     100
#define IOU_THR     0.5f
#define TILE_PTS    64
#define TILE_ELEMS  (TILE_PTS * NUM_CLASSES)   // 5120 floats = 20 KB
#define NUM_TILES   (N_POINTS / TILE_PTS)      // 4096
#define NBINS       4096
#define CAND_CAP    32768
#define BLOCK       256
#define GRID        2048

#ifndef __has_builtin
#define __has_builtin(x) 0
#endif

#if defined(__HIP_DEVICE_COMPILE__) && defined(__gfx1250__) && \
    __has_builtin(__builtin_amdgcn_tensor_load_to_lds) && \
    __has_builtin(__builtin_amdgcn_s_wait_tensorcnt)
#define USE_TDM 1
#else
#define USE_TDM 0
#endif

typedef unsigned int u32x4 __attribute__((ext_vector_type(4)));
typedef int          i32x4 __attribute__((ext_vector_type(4)));
typedef int          i32x8 __attribute__((ext_vector_type(8)));

__device__ __forceinline__ float sigmoidf(float x) {
    return 1.0f / (1.0f + expf(-x));
}

#if USE_TDM
// Build a D# for a contiguous 1-row tile of TILE_ELEMS f32 elements and issue
// TENSOR_LOAD_TO_LDS.  Packing per CDNA5 ISA §8.3/§8.4:
//  group0: [1:0]=count=1, [63:32]=lds_addr, [120:64]=global_addr, [127:126]=2
//  group1: [17:16]=data_size(2 -> 4B), [79:48]=tensor_dim0, [111:80]=tensor_dim1,
//          [127:112]=tile_dim0, [143:128]=tile_dim1, [207:160]=tensor_dim0_stride
__device__ __forceinline__ void tdm_load_tile(unsigned lds_addr, const float* gsrc) {
    unsigned long long ga = (unsigned long long)(uintptr_t)gsrc;
    u32x4 g0;
    g0[0] = 1u;                                            // count=1, user mode
    g0[1] = lds_addr;                                      // LDS byte address
    g0[2] = (unsigned)(ga & 0xFFFFFFFFull);                // global_addr[31:0]
    g0[3] = (unsigned)((ga >> 32) & 0x01FFFFFFull) | 0x80000000u; // [56:32] | type=2
    i32x8 g1;
    g1[0] = (int)(2u << 16);                               // data_size = 4 bytes
    g1[1] = (int)(((unsigned)TILE_ELEMS & 0xFFFFu) << 16); // tensor_dim0[15:0]
    g1[2] = (int)(((unsigned)TILE_ELEMS >> 16) | (1u << 16)); // dim0[31:16], dim1 lo=1
    g1[3] = (int)((unsigned)TILE_ELEMS << 16);             // dim1 hi=0, tile_dim0=5120
    g1[4] = 1;                                             // tile_dim1=1, tile_dim2=0
    g1[5] = TILE_ELEMS;                                    // tensor_dim0_stride lo32
    g1[6] = 0;
    g1[7] = 0;
    i32x4 z4 = {0, 0, 0, 0};
#if __clang_major__ >= 23
    i32x8 z8 = {0, 0, 0, 0, 0, 0, 0, 0};
    __builtin_amdgcn_tensor_load_to_lds(g0, g1, z4, z4, z8, 0);
#else
    __builtin_amdgcn_tensor_load_to_lds(g0, g1, z4, z4, 0);
#endif
}
#endif // USE_TDM

__global__ void init_ws(unsigned* __restrict__ hist, unsigned* __restrict__ counter) {
    int i = blockIdx.x * blockDim.x + threadIdx.x;
    if (i < NBINS) hist[i] = 0u;
    if (i == 0) *counter = 0u;
}

// MODE 0: histogram all 21M scores (top-12-bit float-order bins).
// MODE 1: collect (score, flat_idx) candidates with bin >= *thresh (L2-resident pass).
// Both stream class_preds tiles into LDS with a double-buffered TDM pipeline.
template <int MODE>
__global__ void score_pass(const float* __restrict__ cls,
                           const float* __restrict__ ctr,
                           unsigned* __restrict__ hist,
                           const unsigned* __restrict__ thresh,
                           unsigned* __restrict__ counter,
                           float* __restrict__ cand_s,
                           unsigned* __restrict__ cand_i) {
    __shared__ float    buf[2][TILE_ELEMS];  // 40 KB double buffer
    __shared__ float    sctr[TILE_PTS];
    __shared__ unsigned shist[NBINS];        // 16 KB

    const int tid = threadIdx.x;
    if (MODE == 0) {
        for (int b = tid; b < NBINS; b += BLOCK) shist[b] = 0u;
    }
    unsigned T = 0;
    if (MODE == 1) T = *thresh;

    int tile = blockIdx.x;
    int cur  = 0;
#if USE_TDM
    const bool lead = (tid < 32);            // wave 0 issues DMA (uniform branch)
    if (lead && tile < NUM_TILES)
        tdm_load_tile((unsigned)(uintptr_t)&buf[0][0],
                      cls + (size_t)tile * TILE_ELEMS);
#endif
    while (tile < NUM_TILES) {
        int next = tile + (int)gridDim.x;
#if USE_TDM
        if (lead) {
            if (next < NUM_TILES) {
                // prefetch next tile into the other buffer, wait for current
                tdm_load_tile((unsigned)(uintptr_t)&buf[cur ^ 1][0],
                              cls + (size_t)next * TILE_ELEMS);
                __builtin_amdgcn_s_wait_tensorcnt(1);
            } else {
                __builtin_amdgcn_s_wait_tensorcnt(0);
            }
        }
#else
        {
            const float4* src = (const float4*)(cls + (size_t)tile * TILE_ELEMS);
            float4* dst = (float4*)&buf[cur][0];
            for (int e = tid; e < TILE_ELEMS / 4; e += BLOCK) dst[e] = src[e];
        }
#endif
        if (tid < TILE_PTS)
            sctr[tid] = sigmoidf(ctr[tile * TILE_PTS + tid]);
        __syncthreads();                      // tile data + sctr visible

        const unsigned baseFlat = (unsigned)tile * TILE_ELEMS;
        for (int e = tid; e < TILE_ELEMS; e += BLOCK) {
            float s = sqrtf(sigmoidf(buf[cur][e]) * sctr[e / NUM_CLASSES]);
            unsigned bin = __float_as_uint(s) >> 18;   // monotonic for s > 0
            if (bin >= NBINS) bin = NBINS - 1;
            if (MODE == 0) {
                atomicAdd(&shist[bin], 1u);
            } else if (bin >= T) {
                unsigned p = atomicAdd(counter, 1u);
                if (p < CAND_CAP) { cand_s[p] = s; cand_i[p] = baseFlat + e; }
            }
        }
        __syncthreads();                      // all reads done before buffer reuse
        cur ^= 1;
        tile = next;
    }
    if (MODE == 0) {
        for (int b = tid; b < NBINS; b += BLOCK)
            if (shist[b]) atomicAdd(&hist[b], shist[b]);
    }
}

__global__ void scan_hist(const unsigned* __restrict__ hist,
                          unsigned* __restrict__ thresh) {
    if (threadIdx.x == 0 && blockIdx.x == 0) {
        unsigned cum = 0;
        int b = NBINS - 1;
        for (; b > 0; --b) {
            cum += hist[b];
            if (cum >= MAX_DET) break;
        }
        *thresh = (unsigned)b;
    }
}

__global__ void finalize(const float* __restrict__ cls,
                         const float* __restrict__ ctr,
                         const float* __restrict__ boxp,
                         const float* __restrict__ pts,
                         const float* __restrict__ strd,
                         const unsigned* __restrict__ counter,
                         float* __restrict__ cand_s,         // mutated (consume marks)
                         const unsigned* __restrict__ cand_i,
                         float* __restrict__ out) {
    const int tid = threadIdx.x;
    __shared__ float    rs[BLOCK];
    __shared__ unsigned ri[BLOCK];
    __shared__ int      rp[BLOCK];
    __shared__ unsigned topIdx[MAX_DET];
    __shared__ float    selS[MAX_DET];
    __shared__ float    dbox[MAX_DET][4];
    __shared__ int      selC[MAX_DET];
    __shared__ int      ord[MAX_DET];
    __shared__ int      keep[MAX_DET];

    unsigned cnt = *counter;
    if (cnt > CAND_CAP) cnt = CAND_CAP;

    // Exact top-100 by (score desc, index asc) — matches lax.top_k tie rule.
    for (int k = 0; k < MAX_DET; ++k) {
        float bs = -1.0f; unsigned bi = 0xFFFFFFFFu; int bp = -1;
        for (unsigned p = (unsigned)tid; p < cnt; p += BLOCK) {
            float s = cand_s[p];
            if (s >= 0.0f) {
                unsigned idx = cand_i[p];
                if (s > bs || (s == bs && idx < bi)) { bs = s; bi = idx; bp = (int)p; }
            }
        }
        rs[tid] = bs; ri[tid] = bi; rp[tid] = bp;
        __syncthreads();
        for (int off = BLOCK / 2; off > 0; off >>= 1) {
            if (tid < off) {
                float s2 = rs[tid + off]; unsigned i2 = ri[tid + off];
                if (s2 > rs[tid] || (s2 == rs[tid] && i2 < ri[tid])) {
                    rs[tid] = s2; ri[tid] = i2; rp[tid] = rp[tid + off];
                }
            }
            __syncthreads();
        }
        if (tid == 0) {
            topIdx[k] = (rp[0] >= 0) ? ri[0] : 0u;
            if (rp[0] >= 0) cand_s[rp[0]] = -1.0f;   // consume
        }
        __syncthreads();
    }

    // Gather / decode (faithful to reference: score gathered at POINT index).
    if (tid < MAX_DET) {
        unsigned ti = topIdx[tid];
        unsigned pt = ti / NUM_CLASSES;
        selC[tid]   = (int)(ti % NUM_CLASSES);
        unsigned j  = pt;                        // flat index into [N*C] (ref quirk)
        unsigned gp = j / NUM_CLASSES;
        selS[tid] = sqrtf(sigmoidf(cls[j]) * sigmoidf(ctr[gp]));
        float st = strd[pt];
        float px = pts[pt * 2 + 0], py = pts[pt * 2 + 1];
        float l = boxp[pt * 4 + 0] * st, t = boxp[pt * 4 + 1] * st;
        float r = boxp[pt * 4 + 2] * st, b = boxp[pt * 4 + 3] * st;
        dbox[tid][0] = px - l; dbox[tid][1] = py - t;
        dbox[tid][2] = px + r; dbox[tid][3] = py + b;
        keep[tid] = 1;
    }
    __syncthreads();

    // Stable descending argsort (ties keep original order, like jnp.argsort).
    if (tid < MAX_DET) {
        int rank = 0;
        float s = selS[tid];
        for (int j2 = 0; j2 < MAX_DET; ++j2) {
            float sj = selS[j2];
            if (sj > s || (sj == s && j2 < tid)) ++rank;
        }
        ord[rank] = tid;
    }
    __syncthreads();

    // Greedy NMS in sorted order.
    for (int i = 0; i < MAX_DET; ++i) {
        __syncthreads();
        if (tid < MAX_DET && tid > i && keep[i]) {
            const float* A  = dbox[ord[i]];
            const float* Bx = dbox[ord[tid]];
            float areaA = (A[2] - A[0]) * (A[3] - A[1]);
            float areaB = (Bx[2] - Bx[0]) * (Bx[3] - Bx[1]);
            float lx = fmaxf(A[0], Bx[0]), ly = fmaxf(A[1], Bx[1]);
            float rx = fminf(A[2], Bx[2]), ry = fminf(A[3], Bx[3]);
            float w = fmaxf(rx - lx, 0.0f), h = fmaxf(ry - ly, 0.0f);
            float inter = w * h;
            float iou = inter / (areaA + areaB - inter + 1e-9f);
            if (iou > IOU_THR) keep[tid] = 0;
        }
    }
    __syncthreads();

    if (tid < MAX_DET) {
        int o  = ord[tid];
        int kp = keep[tid];
        for (int c = 0; c < 4; ++c)
            out[tid * 4 + c] = kp ? dbox[o][c] : 0.0f;
        out[4 * MAX_DET + tid] = kp ? selS[o] : 0.0f;
        out[5 * MAX_DET + tid] = kp ? (float)selC[o] : -1.0f;
    }
}

extern "C" void kernel_launch(void* const* d_in, const int* in_sizes, int n_in,
                              void* d_out, int out_size, void* d_ws, size_t ws_size,
                              hipStream_t stream) {
    const float* cls  = (const float*)d_in[0];  // [1, N, 80]
    const float* boxp = (const float*)d_in[1];  // [1, N, 4]
    const float* ctr  = (const float*)d_in[2];  // [1, N, 1]
    const float* pts  = (const float*)d_in[3];  // [N, 2]
    const float* strd = (const float*)d_in[4];  // [N, 1]
    float* out = (float*)d_out;                 // 400 boxes + 100 scores + 100 classes

    char* ws = (char*)d_ws;
    unsigned* hist    = (unsigned*)ws;                       // NBINS * 4 B
    unsigned* counter = (unsigned*)(ws + NBINS * 4);         // 1
    unsigned* thresh  = counter + 1;                          // 1
    float*    cand_s  = (float*)(ws + NBINS * 4 + 256);
    unsigned* cand_i  = (unsigned*)(ws + NBINS * 4 + 256 + CAND_CAP * 4);

    init_ws<<<(NBINS + 255) / 256, 256, 0, stream>>>(hist, counter);

    // 4096 tiles, grid-stride (2 tiles/block), double-buffered TDM
    score_pass<0><<<GRID, BLOCK, 0, stream>>>(cls, ctr, hist, thresh, counter,
                                              cand_s, cand_i);
    scan_hist<<<1, 32, 0, stream>>>(hist, thresh);
    score_pass<1><<<GRID, BLOCK, 0, stream>>>(cls, ctr, hist, thresh, counter,
                                              cand_s, cand_i);
    finalize<<<1, BLOCK, 0, stream>>>(cls, ctr, boxp, pts, strd, counter,
                                      cand_s, cand_i, out);
}